// HallucinationDetector_72018011619385
// MI455X (gfx1250) — compile-verified
//
#include <hip/hip_runtime.h>
#include <math.h>

typedef __attribute__((ext_vector_type(2))) float v2f;
typedef __attribute__((ext_vector_type(8))) float v8f;

#define BB 1024
#define SS 512
#define HH 512
#define DD 2048

__device__ __forceinline__ float sigmoidf_(float x) {
    return 1.0f / (1.0f + __expf(-x));
}

// One block per batch element b. 256 threads = 8 wave32.
// - streams hidden[b] (512x512 f32 = 1MB) through V_WMMA_F32_16X16X4_F32
//   to get per-row logits (B-matrix = probe_w chunk replicated across N).
// - reduces routing_repr[b] (2048 f32) to its L2 norm.
// - thread 0 fuses the 5 risk terms and clamps.
__global__ __launch_bounds__(256) void hallu_risk_kernel(
    const float* __restrict__ routing_entropy,
    const float* __restrict__ moe_confidence,
    const float* __restrict__ memory_mismatch,
    const float* __restrict__ hidden,      // (B,S,H)
    const float* __restrict__ reprm,       // (B,D)
    const float* __restrict__ probe_w,     // (H)
    const float* __restrict__ probe_b,     // (1)
    float* __restrict__ out)               // (B)
{
    __shared__ float s_w[HH];
    __shared__ float s_sem[8];
    __shared__ float s_nrm[8];

    const int b    = blockIdx.x;
    const int tid  = threadIdx.x;
    const int lane = tid & 31;
    const int wave = tid >> 5;

    // stage probe_w into LDS (512 floats = 2KB), coalesced b128 loads
    if (tid < HH / 4) {
        ((float4*)s_w)[tid] = ((const float4*)probe_w)[tid];
    }
    __syncthreads();

    const float pb = probe_b[0];

    // ---- L2 norm of routing_repr row: 2048 floats / 256 threads = 8 each ----
    {
        const float4* rp = (const float4*)(reprm + (size_t)b * DD);
        float4 r0 = rp[tid * 2 + 0];
        float4 r1 = rp[tid * 2 + 1];
        float nrm = r0.x*r0.x + r0.y*r0.y + r0.z*r0.z + r0.w*r0.w
                  + r1.x*r1.x + r1.y*r1.y + r1.z*r1.z + r1.w*r1.w;
        #pragma unroll
        for (int off = 16; off > 0; off >>= 1)
            nrm += __shfl_down(nrm, off, 32);
        if (lane == 0) s_nrm[wave] = nrm;
    }

    // ---- semantic entropy probe via f32 WMMA ----
    // A tile = 16 rows x 4 K of hidden. ISA layout for 16x4 f32 A:
    //   lanes 0-15  hold row (lane&15), K = {0,1} in vgpr {0,1}
    //   lanes 16-31 hold row (lane&15), K = {2,3}
    // => per-lane base pointer pre-offset by hi*2 floats; each WMMA needs
    //    exactly one global_load_b64 per lane (no over-fetch, no selects).
    // B tile = probe_w[k..k+3] replicated across all 16 N columns (LDS
    // broadcast reads), so D[m][n] = partial logit of row m for every n.
    const int row_in_tile = lane & 15;
    const int hi = lane >> 4;            // 0: lanes 0-15, 1: lanes 16-31
    float wave_sem = 0.0f;

    for (int strip = wave; strip < SS / 16; strip += 8) {
        const int row = strip * 16 + row_in_tile;
        const float* abase = hidden + ((size_t)b * SS + row) * HH + hi * 2;

        v8f acc0 = {0.f,0.f,0.f,0.f,0.f,0.f,0.f,0.f};
        v8f acc1 = {0.f,0.f,0.f,0.f,0.f,0.f,0.f,0.f};

        #pragma unroll 8
        for (int kb = 0; kb < HH / 4; kb += 2) {
            // exact 8B A-fragments: one b64 load per WMMA per lane
            v2f af0 = *(const v2f*)(abase + kb * 4);
            v2f af1 = *(const v2f*)(abase + kb * 4 + 4);

            const float* wp0 = s_w + kb * 4 + hi * 2;   // LDS broadcast
            v2f bf0 = {wp0[0], wp0[1]};
            v2f bf1 = {wp0[4], wp0[5]};

            acc0 = __builtin_amdgcn_wmma_f32_16x16x4_f32(
                false, af0, false, bf0, (short)0, acc0, false, false);
            acc1 = __builtin_amdgcn_wmma_f32_16x16x4_f32(
                false, af1, false, bf1, (short)0, acc1, false, false);
        }

        // lane n<16 holds rows strip*16+0..7 in acc[0..7]; lanes>=16 rows +8..15
        float s = 0.0f;
        #pragma unroll
        for (int j = 0; j < 8; ++j)
            s += sigmoidf_(acc0[j] + acc1[j] + pb);
        // strip total = rows 0-7 part (any lane<16) + rows 8-15 part (any lane>=16)
        wave_sem += __shfl(s, 0, 32) + __shfl(s, 16, 32);
    }
    if (lane == 0) s_sem[wave] = wave_sem;
    __syncthreads();

    if (tid == 0) {
        float sem = 0.0f, nn = 0.0f;
        #pragma unroll
        for (int i = 0; i < 8; ++i) { sem += s_sem[i]; nn += s_nrm[i]; }
        sem *= (1.0f / (float)SS);

        float sv    = sqrtf(nn);
        float eig   = 1.0f / (sv + 1e-8f);
        float neig  = sigmoidf_(eig - 1.0f);
        float ne    = routing_entropy[b] * 0.48089834696298783f; // 1/ln(8)
        float ic    = 1.0f - moe_confidence[b];
        float nm    = sigmoidf_(memory_mismatch[b] - 2.0f);

        float risk = 0.25f * ne - 0.2f * ic + 0.2f * nm
                   + 0.2f * sem + 0.15f * neig;
        out[b] = fminf(fmaxf(risk, 0.0f), 1.0f);
    }
}

extern "C" void kernel_launch(void* const* d_in, const int* in_sizes, int n_in,
                              void* d_out, int out_size, void* d_ws, size_t ws_size,
                              hipStream_t stream) {
    (void)in_sizes; (void)n_in; (void)d_ws; (void)ws_size; (void)out_size;
    const float* routing_entropy = (const float*)d_in[0];
    const float* moe_confidence  = (const float*)d_in[1];
    const float* memory_mismatch = (const float*)d_in[2];
    // d_in[3] = memory_loss: unused by the reference computation
    const float* hidden          = (const float*)d_in[4];
    const float* reprm           = (const float*)d_in[5];
    const float* probe_w         = (const float*)d_in[6];
    const float* probe_b         = (const float*)d_in[7];
    float* out                   = (float*)d_out;

    hallu_risk_kernel<<<dim3(BB), dim3(256), 0, stream>>>(
        routing_entropy, moe_confidence, memory_mismatch,
        hidden, reprm, probe_w, probe_b, out);
}